// UniversalHMM_70128226009812
// MI455X (gfx1250) — compile-verified
//
#include <hip/hip_runtime.h>
#include <hip/hip_bf16.h>
#include <math.h>

// Problem constants
#define BB 4096
#define TT 512
#define SS 6
#define FF 12
#define EPS 1e-8f

// Workspace float-offset layout (header kept tiny, then big arrays)
#define WF_OFF     0            // 24 x 16 weight matrix (cols 6..15 zero)
#define CTERM_OFF  384          // 16 per-state constants (6 used)
#define LTRANS_OFF 400          // 6x6 log transition
#define LINIT_OFF  436          // 6 log initial
#define HDR_FLOATS 512
#define EMIS_OFF   HDR_FLOATS                      // [T][B][S] floats = 12,582,912
#define EMIS_FLOATS ((size_t)TT * BB * SS)
#define BP_BYTE_OFF ((size_t)(HDR_FLOATS + EMIS_FLOATS) * 4)  // (T-1)*B u32, 3b/state packed

typedef float v2f __attribute__((ext_vector_type(2)));
typedef float v4f __attribute__((ext_vector_type(4)));
typedef float v8f __attribute__((ext_vector_type(8)));

// ---------------------------------------------------------------------------
// Kernel 0: parameter preprocessing (tiny). Builds:
//   Wf[j][n]  j<12:  means[n][j]*inv2[n][j]      (coef of obs)
//             j>=12: -0.5*inv2[n][j-12]          (coef of obs^2)
//   cterm[n] = -0.5*sum_f means^2*inv2 - sum_f log(scales)
//   ltrans   = log(softmax(transition_logits,axis=-1)+EPS)
//   linit    = log(softmax(initial_logits)+EPS)
// ---------------------------------------------------------------------------
__global__ void hmm_prep_kernel(const float* __restrict__ trans_logits,
                                const float* __restrict__ init_logits,
                                const float* __restrict__ means,
                                const float* __restrict__ log_scales,
                                float* __restrict__ hdr) {
    int tid = threadIdx.x;
    for (int i = tid; i < HDR_FLOATS; i += blockDim.x) hdr[i] = 0.0f;
    __syncthreads();

    // Wf entries: one thread per (s,f)
    if (tid < SS * FF) {
        int s = tid / FF, f = tid % FF;
        float x = log_scales[s * FF + f];
        float scale = log1pf(expf(x)) + 1e-6f;   // softplus + 1e-6
        float denom = scale + 1e-6f;
        float inv2 = 1.0f / (denom * denom);
        hdr[WF_OFF + f * 16 + s]        = means[s * FF + f] * inv2;
        hdr[WF_OFF + (12 + f) * 16 + s] = -0.5f * inv2;
    }
    // cterm: one thread per state
    if (tid < SS) {
        int s = tid;
        float quad = 0.0f, logdet = 0.0f;
        for (int f = 0; f < FF; ++f) {
            float x = log_scales[s * FF + f];
            float scale = log1pf(expf(x)) + 1e-6f;
            float denom = scale + 1e-6f;
            float inv2 = 1.0f / (denom * denom);
            float m = means[s * FF + f];
            quad += m * m * inv2;
            logdet += logf(scale);
        }
        hdr[CTERM_OFF + s] = -0.5f * quad - logdet;
    }
    // log transition rows: one thread per prev-state
    if (tid < SS) {
        int sp = tid;
        float mx = trans_logits[sp * SS + 0];
        for (int s = 1; s < SS; ++s) mx = fmaxf(mx, trans_logits[sp * SS + s]);
        float sum = 0.0f;
        for (int s = 0; s < SS; ++s) sum += expf(trans_logits[sp * SS + s] - mx);
        for (int s = 0; s < SS; ++s)
            hdr[LTRANS_OFF + sp * SS + s] =
                logf(expf(trans_logits[sp * SS + s] - mx) / sum + EPS);
    }
    // log initial
    if (tid == 0) {
        float mx = init_logits[0];
        for (int s = 1; s < SS; ++s) mx = fmaxf(mx, init_logits[s]);
        float sum = 0.0f;
        for (int s = 0; s < SS; ++s) sum += expf(init_logits[s] - mx);
        for (int s = 0; s < SS; ++s)
            hdr[LINIT_OFF + s] = logf(expf(init_logits[s] - mx) / sum + EPS);
    }
}

// ---------------------------------------------------------------------------
// Kernel 1: emission log-probs via V_WMMA_F32_16X16X4_F32.
// GEMM: [B*T rows x K=24] x [24 x N=16 (6 used)], K split into 6 chunks of 4.
// A layout (ISA 7.12.2, 32-bit A 16x4): element (m,k) in VGPR (k&1),
// lane (k>>1)*16 + m.  B layout symmetric: (k,n) in VGPR (k&1), lane
// (k>>1)*16 + n.  C/D: VGPR r holds row M = r + 8*(lane>>4), col N = lane&15.
//
// A-fragments are selected straight from the vector load registers with
// explicit scalar ternaries (no local array -> no dynamic-index cndmask
// forest), and the obs^2 fragments (K chunks 3..5) are just the squares of
// the already-selected obs fragments (K chunks 0..2).
// Output stored transposed [T][B][S] so the recursion reads coalesced.
// ---------------------------------------------------------------------------
__global__ void hmm_emission_wmma_kernel(const float* __restrict__ obs,
                                         const float* __restrict__ hdr,
                                         float* __restrict__ ews) {
    const int lane = threadIdx.x & 31;
    const int waveInBlk = threadIdx.x >> 5;
    const int n = lane & 15;          // output state column (valid if < 6)
    const int half = lane >> 4;       // lane half selects K sub-pair / M group
    const bool h1 = (half != 0);

    // Preload B fragments (K=24 weights) and accumulator seed, once per wave.
    v2f bfrag[6];
    #pragma unroll
    for (int kc = 0; kc < 6; ++kc) {
        int kb = 4 * kc + 2 * half;
        bfrag[kc].x = hdr[WF_OFF + kb * 16 + n];
        bfrag[kc].y = hdr[WF_OFF + (kb + 1) * 16 + n];
    }
    const float ct = hdr[CTERM_OFF + n];   // zero for n >= 6

    const int numTiles = (BB * TT) / 16;               // 131072
    const int waveId = blockIdx.x * (blockDim.x >> 5) + waveInBlk;
    const int waveStride = gridDim.x * (blockDim.x >> 5);

    for (int tile = waveId; tile < numTiles; tile += waveStride) {
        const int btBase = tile * 16;
        const int m = lane & 15;                        // A-matrix row in tile
        const float* orow = obs + (size_t)(btBase + m) * FF;  // 48B, 16B aligned
        v4f p0 = __builtin_nontemporal_load((const v4f*)(orow));
        v4f p1 = __builtin_nontemporal_load((const v4f*)(orow + 4));
        v4f p2 = __builtin_nontemporal_load((const v4f*)(orow + 8));

        // obs fragments (K chunks 0..2): low half takes (x,y), high half (z,w)
        v2f a0, a1, a2;
        a0.x = h1 ? p0.z : p0.x;  a0.y = h1 ? p0.w : p0.y;
        a1.x = h1 ? p1.z : p1.x;  a1.y = h1 ? p1.w : p1.y;
        a2.x = h1 ? p2.z : p2.x;  a2.y = h1 ? p2.w : p2.y;
        // obs^2 fragments (K chunks 3..5) = element-wise squares of the above
        v2f a3 = a0 * a0;
        v2f a4 = a1 * a1;
        v2f a5 = a2 * a2;

        v8f acc = {ct, ct, ct, ct, ct, ct, ct, ct};
        acc = __builtin_amdgcn_wmma_f32_16x16x4_f32(false, a0, false, bfrag[0], (short)0, acc, false, false);
        acc = __builtin_amdgcn_wmma_f32_16x16x4_f32(false, a1, false, bfrag[1], (short)0, acc, false, false);
        acc = __builtin_amdgcn_wmma_f32_16x16x4_f32(false, a2, false, bfrag[2], (short)0, acc, false, false);
        acc = __builtin_amdgcn_wmma_f32_16x16x4_f32(false, a3, false, bfrag[3], (short)0, acc, false, false);
        acc = __builtin_amdgcn_wmma_f32_16x16x4_f32(false, a4, false, bfrag[4], (short)0, acc, false, false);
        acc = __builtin_amdgcn_wmma_f32_16x16x4_f32(false, a5, false, bfrag[5], (short)0, acc, false, false);

        if (n < SS) {
            #pragma unroll
            for (int r = 0; r < 8; ++r) {
                int bt = btBase + r + 8 * half;         // global row
                int b = bt >> 9;                        // /T (T=512)
                int t = bt & (TT - 1);
                ews[((size_t)t * BB + b) * SS + n] = acc[r];
            }
        }
    }
}

// ---------------------------------------------------------------------------
// Kernel 2: Viterbi forward + backtrace, one thread per batch element.
// Backpointers packed 3 bits/state -> one u32 per (b,t): coalesced, L2-resident.
// Strict '>' updates reproduce jnp.argmax first-max tie-breaking.
// ---------------------------------------------------------------------------
__global__ void hmm_viterbi_kernel(const float* __restrict__ hdr,
                                   const float* __restrict__ ews,
                                   unsigned int* __restrict__ bpws,
                                   float* __restrict__ out) {
    const int b = blockIdx.x * blockDim.x + threadIdx.x;
    if (b >= BB) return;

    float lt[SS * SS];
    #pragma unroll
    for (int i = 0; i < SS * SS; ++i) lt[i] = hdr[LTRANS_OFF + i];

    float v[SS];
    {
        const float* e0 = ews + (size_t)b * SS;   // t = 0
        #pragma unroll
        for (int s = 0; s < SS; ++s) v[s] = hdr[LINIT_OFF + s] + e0[s];
    }

    for (int t = 1; t < TT; ++t) {
        const float* e = ews + ((size_t)t * BB + b) * SS;
        float e0 = e[0], e1 = e[1], e2 = e[2], e3 = e[3], e4 = e[4], e5 = e[5];
        float nv[SS];
        unsigned int packed = 0u;
        #pragma unroll
        for (int s = 0; s < SS; ++s) {
            float best = v[0] + lt[0 * SS + s];
            int bpi = 0;
            #pragma unroll
            for (int sp = 1; sp < SS; ++sp) {
                float cand = v[sp] + lt[sp * SS + s];
                bool better = cand > best;             // strict: keep first max
                best = better ? cand : best;
                bpi = better ? sp : bpi;
            }
            float ee = (s == 0) ? e0 : (s == 1) ? e1 : (s == 2) ? e2
                     : (s == 3) ? e3 : (s == 4) ? e4 : e5;
            nv[s] = best + ee;
            packed |= ((unsigned int)bpi) << (3 * s);
        }
        #pragma unroll
        for (int s = 0; s < SS; ++s) v[s] = nv[s];
        bpws[(size_t)(t - 1) * BB + b] = packed;
    }

    // Final argmax + score
    float bestVal = v[0];
    int last = 0;
    #pragma unroll
    for (int s = 1; s < SS; ++s) {
        bool better = v[s] > bestVal;
        bestVal = better ? v[s] : bestVal;
        last = better ? s : last;
    }
    __builtin_nontemporal_store(bestVal, out + (size_t)BB * TT + b);  // score

    // Backtrace (path stored as float casts of int states)
    int cur = last;
    __builtin_nontemporal_store((float)cur, out + (size_t)b * TT + (TT - 1));
    for (int t = TT - 2; t >= 0; --t) {
        unsigned int packed = bpws[(size_t)t * BB + b];
        cur = (int)((packed >> (3 * cur)) & 7u);
        __builtin_nontemporal_store((float)cur, out + (size_t)b * TT + t);
    }
}

// ---------------------------------------------------------------------------
extern "C" void kernel_launch(void* const* d_in, const int* in_sizes, int n_in,
                              void* d_out, int out_size, void* d_ws, size_t ws_size,
                              hipStream_t stream) {
    const float* obs        = (const float*)d_in[0];  // [B,T,F]
    const float* trans      = (const float*)d_in[1];  // [S,S]
    const float* initl      = (const float*)d_in[2];  // [S]
    const float* means      = (const float*)d_in[3];  // [S,F]
    const float* log_scales = (const float*)d_in[4];  // [S,F]
    float* out = (float*)d_out;                       // [B*T] path + [B] score

    float* hdr = (float*)d_ws;
    float* ews = hdr + EMIS_OFF;
    unsigned int* bpws = (unsigned int*)((char*)d_ws + BP_BYTE_OFF);

    hmm_prep_kernel<<<1, 128, 0, stream>>>(trans, initl, means, log_scales, hdr);
    // 2048 blocks x 8 waves = 16384 waves; 8 tiles each (131072 tiles total)
    hmm_emission_wmma_kernel<<<2048, 256, 0, stream>>>(obs, hdr, ews);
    hmm_viterbi_kernel<<<BB / 256, 256, 0, stream>>>(hdr, ews, bpws, out);
}